// LayerNorm_v2_21741124452866
// MI455X (gfx1250) — compile-verified
//
#include <hip/hip_runtime.h>
#include <cstdint>

// LayerNorm-with-scan reference:
//   stats from x[0] only, serial scan over D per sequence position s:
//     mean = (mean + d) / i ;  var += (d - mean)^2 ; i = 1..D ; var /= (D-1)
//   out[b,s,d] = alpha[d] * (x[b,s,d] - mean[s]) / sqrt(var[s]+eps) + beta[d]
//
// Memory-bound: ~272 MiB total -> ~11.5us floor @ 23.3 TB/s.
// Pass 1 uses CDNA5 async global->LDS DMA (ASYNCcnt) with double buffering.

#define EPS_F 1e-5f

constexpr int Bsz = 8;
constexpr int S   = 4096;
constexpr int D   = 1024;

constexpr int ROWS   = 32;          // rows (s values) per block == lanes per wave32
constexpr int CHUNK  = 128;         // floats of D staged per LDS tile (512 B per row)
constexpr int STRIDE = CHUNK + 4;   // 132 floats: 528 B pitch, 16B-aligned, ~conflict-free
constexpr int NCHUNK = D / CHUNK;   // 8

// ---------------- Pass 1: per-s serial scan with async-LDS double buffering ----------
__global__ __launch_bounds__(ROWS) void ln_stats_kernel(const float* __restrict__ x,
                                                        float* __restrict__ meanOut,
                                                        float* __restrict__ rstdOut) {
  __shared__ float smem[2 * ROWS * STRIDE];   // 33,792 bytes
  const int lane    = threadIdx.x;            // 0..31, one s-row per lane
  const int rowBase = blockIdx.x * ROWS;
  const uint32_t ldsBase = (uint32_t)(uintptr_t)&smem[0]; // low 32 bits == LDS offset

  // Issue one 32-row x CHUNK tile: 32 async b128 instructions, each moving a
  // fully-coalesced 512-B row-chunk (32 lanes x 16 B).
  auto issue_fill = [&](int chunk, int parity) {
    const float* g0 = x + (size_t)rowBase * D + chunk * CHUNK + lane * 4;
    uint32_t l0 = ldsBase + (uint32_t)parity * (ROWS * STRIDE * 4) + (uint32_t)lane * 16u;
#pragma unroll
    for (int r = 0; r < ROWS; ++r) {
      const float* g = g0 + (size_t)r * D;
      uint32_t l = l0 + (uint32_t)r * (STRIDE * 4);
      asm volatile("global_load_async_to_lds_b128 %0, %1, off"
                   :: "v"(l), "v"(g) : "memory");
    }
  };

  float mean = 0.0f, var = 0.0f, fi = 1.0f;
  issue_fill(0, 0);
  for (int c = 0; c < NCHUNK; ++c) {
    if (c + 1 < NCHUNK) {
      issue_fill(c + 1, (c + 1) & 1);
      // async loads complete in order: <=32 outstanding => previous tile landed
      asm volatile("s_wait_asynccnt 32" ::: "memory");
    } else {
      asm volatile("s_wait_asynccnt 0" ::: "memory");
    }
    const float* row = &smem[((size_t)(c & 1) * ROWS + lane) * STRIDE];
#pragma unroll 8
    for (int dd = 0; dd < CHUNK; ++dd) {
      float d = row[dd];
      mean = (mean + d) / fi;          // IEEE divide: match reference rounding
      float t = d - mean;
      var = fmaf(t, t, var);
      fi += 1.0f;
    }
  }
  var = var / (float)(D - 1);
  const int s = rowBase + lane;
  meanOut[s] = mean;
  rstdOut[s] = 1.0f / sqrtf(var + EPS_F);
}

// ---------------- Pass 2: streaming normalize, one (b,s) row per block --------------
__global__ __launch_bounds__(256) void ln_apply_kernel(const float4* __restrict__ x4,
                                                       const float4* __restrict__ a4,
                                                       const float4* __restrict__ b4,
                                                       const float* __restrict__ meanArr,
                                                       const float* __restrict__ rstdArr,
                                                       float4* __restrict__ out4) {
  const int tid = threadIdx.x;
  const int blk = blockIdx.x;           // one row of D=1024 floats (256 thr x float4)
  const int s   = blk & (S - 1);        // uniform per block -> scalar loads
  const float m = meanArr[s];
  const float r = rstdArr[s];
  const size_t base = (size_t)blk * (D / 4);
  float4 xv = x4[base + tid];
  float4 av = a4[tid];
  float4 bv = b4[tid];
  float4 o;
  o.x = fmaf(av.x, (xv.x - m) * r, bv.x);
  o.y = fmaf(av.y, (xv.y - m) * r, bv.y);
  o.z = fmaf(av.z, (xv.z - m) * r, bv.z);
  o.w = fmaf(av.w, (xv.w - m) * r, bv.w);
  out4[base + tid] = o;
}

extern "C" void kernel_launch(void* const* d_in, const int* in_sizes, int n_in,
                              void* d_out, int out_size, void* d_ws, size_t ws_size,
                              hipStream_t stream) {
  const float* x     = (const float*)d_in[0];
  const float* alpha = (const float*)d_in[1];
  const float* beta  = (const float*)d_in[2];
  float* out = (float*)d_out;

  float* meanArr = (float*)d_ws;       // S floats
  float* rstdArr = meanArr + S;        // S floats  (32 KiB total scratch)

  ln_stats_kernel<<<S / ROWS, ROWS, 0, stream>>>(x, meanArr, rstdArr);
  ln_apply_kernel<<<Bsz * S, 256, 0, stream>>>((const float4*)x,
                                               (const float4*)alpha,
                                               (const float4*)beta,
                                               meanArr, rstdArr,
                                               (float4*)out);
}